// QMessagePassing_38663295598907
// MI455X (gfx1250) — compile-verified
//
#include <hip/hip_runtime.h>

#define N_NODES 50000
#define N_EDGES 800000
#define FDIM 16
#define TFEAT 64   // 4 quaternion components * FDIM

typedef __attribute__((ext_vector_type(2))) float v2f;
typedef __attribute__((ext_vector_type(8))) float v8f;

// ---------------------------------------------------------------------------
// float atomic max via signed/unsigned bit trick (works for all finite + -inf)
// ---------------------------------------------------------------------------
__device__ __forceinline__ void atomic_max_f32(float* addr, float val) {
  if (val >= 0.0f) {
    atomicMax((int*)addr, __float_as_int(val));
  } else {
    atomicMin((unsigned int*)addr, (unsigned int)__float_as_int(val));
  }
}

// ---------------------------------------------------------------------------
// Kernel 0: init workspace (zmax=-inf, denom=0, num=0). d_ws is poisoned by
// the harness, so this must run every launch.
// ---------------------------------------------------------------------------
__global__ void qmp_init_ws(float* __restrict__ zmax, float* __restrict__ denom,
                            float* __restrict__ num) {
  int idx = blockIdx.x * blockDim.x + threadIdx.x;
  if (idx < N_NODES * TFEAT) {
    zmax[idx]  = -__builtin_inff();
    denom[idx] = 0.0f;
    num[idx]   = 0.0f;
  }
}

// ---------------------------------------------------------------------------
// Kernel 1: per-edge segment max.  One thread per (edge, feature) pair,
// looping the 4 quaternion components (coalesced 4B/lane over edge_attr).
// ---------------------------------------------------------------------------
__global__ void qmp_edge_max(const float* __restrict__ q,
                             const float* __restrict__ ea,
                             const int* __restrict__ eidx,
                             const float* __restrict__ beta,
                             float* __restrict__ zmax) {
  int gid = blockIdx.x * blockDim.x + threadIdx.x;   // over E*F
  if (gid >= N_EDGES * FDIM) return;
  int e = gid >> 4;
  int f = gid & 15;
  int src = eidx[e];
  int dst = eidx[N_EDGES + e];
  float b = beta[0];
#pragma unroll
  for (int c = 0; c < 4; ++c) {
    float m = q[(c * N_NODES + src) * FDIM + f] + ea[(c * N_EDGES + e) * FDIM + f];
    atomic_max_f32(&zmax[dst * TFEAT + c * FDIM + f], m * b);
  }
}

// ---------------------------------------------------------------------------
// Kernel 2: per-edge exp + accumulate numerator/denominator.
// agg = (sum_e msg*exp(z-zmax)) / (sum_e exp(z-zmax))
// ---------------------------------------------------------------------------
__global__ void qmp_edge_accum(const float* __restrict__ q,
                               const float* __restrict__ ea,
                               const int* __restrict__ eidx,
                               const float* __restrict__ beta,
                               const float* __restrict__ zmax,
                               float* __restrict__ denom,
                               float* __restrict__ num) {
  int gid = blockIdx.x * blockDim.x + threadIdx.x;
  if (gid >= N_EDGES * FDIM) return;
  int e = gid >> 4;
  int f = gid & 15;
  int src = eidx[e];
  int dst = eidx[N_EDGES + e];
  float b = beta[0];
#pragma unroll
  for (int c = 0; c < 4; ++c) {
    int t = c * FDIM + f;
    float m  = q[(c * N_NODES + src) * FDIM + f] + ea[(c * N_EDGES + e) * FDIM + f];
    float ez = __expf(m * b - zmax[dst * TFEAT + t]);
    atomicAdd(&denom[dst * TFEAT + t], ez);
    atomicAdd(&num[dst * TFEAT + t], m * ez);
  }
}

// ---------------------------------------------------------------------------
// Kernel 3: node pass.  One wave (32 lanes) per 16-node tile; x = q + agg
// staged in LDS, then two quaternion-linear layers via V_WMMA_F32_16X16X4_F32
// (K=16 done as 4 chunks of K=4; Hamilton minus-signs folded into B).
// 50000 nodes = 3125 tiles exactly -> EXEC always all-ones (WMMA requirement).
// ---------------------------------------------------------------------------
__global__ void __launch_bounds__(32)
qmp_node_mlp(const float* __restrict__ q,
             const float* __restrict__ W1, const float* __restrict__ b1,
             const float* __restrict__ W2, const float* __restrict__ b2,
             const float* __restrict__ num, const float* __restrict__ denom,
             float* __restrict__ out) {
  __shared__ float tile[4 * 16 * FDIM];   // [c][M][f], M = node-in-tile

  const int lane = threadIdx.x;           // 0..31 (wave32)
  const int m0   = blockIdx.x * 16;       // node base of this tile
  const int hi   = lane >> 4;             // 0 for lanes 0-15, 1 for 16-31
  const int n    = lane & 15;

  // ---- Phase 1: x = q + agg  into LDS (coalesced over num/denom) ----------
#pragma unroll
  for (int j = 0; j < 32; ++j) {
    int t = j * 32 + lane;                // flat over [M][c][f] (node-major)
    int M = t >> 6;
    int c = (t >> 4) & 3;
    int f = t & 15;
    float dn  = denom[m0 * TFEAT + t];
    float nm  = num[m0 * TFEAT + t];
    float agg = (dn > 0.0f) ? (nm / dn) : 0.0f;   // empty segment -> 0
    tile[(c * 16 + M) * FDIM + f] = q[(c * N_NODES + (m0 + M)) * FDIM + f] + agg;
  }

  // Hamilton product tables: out o = sum_ac  sgn * X[ac] @ W[wcomp]
  const int   wcomp[4][4] = {{0,1,2,3},{1,0,3,2},{2,3,0,1},{3,2,1,0}};
  const float sgn[4][4]   = {{1.f,-1.f,-1.f,-1.f},{1.f,1.f,1.f,-1.f},
                             {1.f,-1.f,1.f,1.f},{1.f,1.f,-1.f,1.f}};

#pragma unroll
  for (int layer = 0; layer < 2; ++layer) {
    const float* W    = (layer == 0) ? W1 : W2;
    const float* bias = (layer == 0) ? b1 : b2;

    // A fragments (16x4 f32): lane holds row M=n; VGPR0={K0|K2}, VGPR1={K1|K3}
    v2f a[4][4];
#pragma unroll
    for (int c = 0; c < 4; ++c)
#pragma unroll
      for (int kc = 0; kc < 4; ++kc) {
        int base = (c * 16 + n) * FDIM + 4 * kc + 2 * hi;
        a[c][kc].x = tile[base];
        a[c][kc].y = tile[base + 1];
      }

    v8f acc[4];
#pragma unroll
    for (int o = 0; o < 4; ++o) {
      v8f c8 = {0.f, 0.f, 0.f, 0.f, 0.f, 0.f, 0.f, 0.f};
#pragma unroll
      for (int ac = 0; ac < 4; ++ac) {
        const int   wc = wcomp[o][ac];
        const float s  = sgn[o][ac];
#pragma unroll
        for (int kc = 0; kc < 4; ++kc) {
          // B fragment (4x16 f32): lane holds col N=n; same lo/hi K split as A
          v2f bfr;
          bfr.x = s * W[(wc * FDIM + 4 * kc + 2 * hi) * FDIM + n];
          bfr.y = s * W[(wc * FDIM + 4 * kc + 2 * hi + 1) * FDIM + n];
          c8 = __builtin_amdgcn_wmma_f32_16x16x4_f32(
              /*neg_a=*/false, a[ac][kc], /*neg_b=*/false, bfr,
              /*c_mod=*/(short)0, c8, /*reuse_a=*/false, /*reuse_b=*/false);
        }
      }
      float bv = bias[o * FDIM + n];
#pragma unroll
      for (int v = 0; v < 8; ++v) {
        float x = c8[v] + bv;
        if (layer == 0) x = fmaxf(x, 0.0f);   // relu after layer 1 only
        c8[v] = x;
      }
      acc[o] = c8;
    }

    // C/D layout -> LDS [c][M][f]: VGPR v holds row (hi*8+v), col n
#pragma unroll
    for (int o = 0; o < 4; ++o)
#pragma unroll
      for (int v = 0; v < 8; ++v)
        tile[(o * 16 + (hi * 8 + v)) * FDIM + n] = acc[o][v];
  }

  // ---- Phase 3: coalesced store out[c][m0+M][f] ---------------------------
#pragma unroll
  for (int j = 0; j < 32; ++j) {
    int t = j * 32 + lane;        // flat over [c][M][f] == tile layout
    int c = t >> 8;
    int r = t & 255;
    out[c * (N_NODES * FDIM) + m0 * FDIM + r] = tile[t];
  }
}

// ---------------------------------------------------------------------------
extern "C" void kernel_launch(void* const* d_in, const int* in_sizes, int n_in,
                              void* d_out, int out_size, void* d_ws, size_t ws_size,
                              hipStream_t stream) {
  const float* q    = (const float*)d_in[0];   // [4,N,F]
  const float* ea   = (const float*)d_in[1];   // [4,E,F]
  const int*   eidx = (const int*)d_in[2];     // [2,E]
  const float* W1   = (const float*)d_in[3];   // [4,F,F]
  const float* b1   = (const float*)d_in[4];   // [4,F]
  const float* W2   = (const float*)d_in[5];   // [4,F,F]
  const float* b2   = (const float*)d_in[6];   // [4,F]
  const float* beta = (const float*)d_in[7];   // [1]
  float* out = (float*)d_out;                  // [4,N,F]

  // workspace: zmax | denom | num, each [N,64] f32  (38.4 MB total)
  float* zmax  = (float*)d_ws;
  float* denom = zmax + (size_t)N_NODES * TFEAT;
  float* num   = denom + (size_t)N_NODES * TFEAT;

  { // init
    int n = N_NODES * TFEAT;
    qmp_init_ws<<<(n + 255) / 256, 256, 0, stream>>>(zmax, denom, num);
  }
  { // edge passes
    int n = N_EDGES * FDIM;
    qmp_edge_max<<<(n + 255) / 256, 256, 0, stream>>>(q, ea, eidx, beta, zmax);
    qmp_edge_accum<<<(n + 255) / 256, 256, 0, stream>>>(q, ea, eidx, beta, zmax,
                                                        denom, num);
  }
  // node MLP: one wave per 16-node tile, 3125 tiles exactly
  qmp_node_mlp<<<N_NODES / 16, 32, 0, stream>>>(q, W1, b1, W2, b2, num, denom, out);
}